// MaskedDiceLoss_SliceConsist_77738908057882
// MI455X (gfx1250) — compile-verified
//
#include <hip/hip_runtime.h>

#define SMOOTH 0.1f

typedef __attribute__((ext_vector_type(2))) float v2f;
typedef __attribute__((ext_vector_type(8))) float v8f;

__device__ __forceinline__ float sigmoidf_fast(float x) {
  return 1.0f / (1.0f + __expf(-x));
}

// Exact reduction of 256 f32 values staged in LDS, using
// V_WMMA_F32_16X16X4_F32:  D = A(16x4) * ones(4x16) + C  => per-row sums.
// B = all-ones is layout-invariant; A uses the documented 32-bit 16x4 layout
// (lanes 0-15 hold K=0,1 in V0,V1; lanes 16-31 hold K=2,3).
// Must be called by one fully-active wave32. Returns the total in all lanes.
__device__ __forceinline__ float wmma_reduce256(const float* lds, int lane) {
  v2f ones;
  ones[0] = 1.0f;
  ones[1] = 1.0f;
  v8f acc = {};
  const int m  = lane & 15;
  const int kh = (lane >> 4) << 1;  // 0 for lanes 0-15, 2 for lanes 16-31
#pragma unroll
  for (int t = 0; t < 4; ++t) {
    v2f a;
    a[0] = lds[t * 64 + m * 4 + kh + 0];
    a[1] = lds[t * 64 + m * 4 + kh + 1];
    acc = __builtin_amdgcn_wmma_f32_16x16x4_f32(false, a, false, ones,
                                                (short)0, acc, false, false);
  }
  // C/D layout: VGPR r, lanes 0-15 -> M=r ; lanes 16-31 -> M=r+8 (all N equal)
  float s = acc[0] + acc[1] + acc[2] + acc[3] + acc[4] + acc[5] + acc[6] + acc[7];
  s += __shfl_xor(s, 16, 32);  // combine M=0..7 with M=8..15 halves
  return s;
}

__global__ void __launch_bounds__(256) mdl_init_kernel(float* __restrict__ ws) {
  ws[blockIdx.x * 256 + threadIdx.x] = 0.0f;
}

// Grid: (32 chunks, 64 slices, 8 batches), 256 threads.
// Each block handles 2048 elements of one (b,s) slice: two float4 loads/thread.
__global__ void __launch_bounds__(256) mdl_partial_kernel(
    const float* __restrict__ pred, const float* __restrict__ gt,
    float* __restrict__ acc_pg, float* __restrict__ acc_p,
    float* __restrict__ acc_g, float* __restrict__ acc_d2) {
  const int tid   = threadIdx.x;
  const int chunk = blockIdx.x;  // 0..31
  const int s     = blockIdx.y;  // 0..63
  const int b     = blockIdx.z;  // 0..7
  const int slice = b * 64 + s;
  const long base = (long)slice * 65536 + (long)chunk * 2048;
  const bool have_next = (s < 63);

  float spg = 0.0f, sp = 0.0f, sg = 0.0f, sd2 = 0.0f;
#pragma unroll
  for (int half = 0; half < 2; ++half) {
    const long off = base + half * 1024 + tid * 4;
    const float4 pv = *(const float4*)(pred + off);
    const float4 gv = *(const float4*)(gt + off);
    const float p0 = sigmoidf_fast(pv.x);
    const float p1 = sigmoidf_fast(pv.y);
    const float p2 = sigmoidf_fast(pv.z);
    const float p3 = sigmoidf_fast(pv.w);
    spg += p0 * gv.x + p1 * gv.y + p2 * gv.z + p3 * gv.w;
    sp  += p0 + p1 + p2 + p3;
    sg  += gv.x + gv.y + gv.z + gv.w;
    if (have_next) {
      const float4 nv = *(const float4*)(pred + off + 65536);
      const float d0 = p0 - sigmoidf_fast(nv.x);
      const float d1 = p1 - sigmoidf_fast(nv.y);
      const float d2 = p2 - sigmoidf_fast(nv.z);
      const float d3 = p3 - sigmoidf_fast(nv.w);
      sd2 += d0 * d0 + d1 * d1 + d2 * d2 + d3 * d3;
    }
  }

  __shared__ float lds_pg[256];
  __shared__ float lds_p[256];
  __shared__ float lds_g[256];
  __shared__ float lds_d2[256];
  lds_pg[tid] = spg;
  lds_p[tid]  = sp;
  lds_g[tid]  = sg;
  lds_d2[tid] = sd2;
  __syncthreads();

  if (tid < 32) {  // wave 0 only: EXEC all-ones within the wave (WMMA req.)
    const float tpg = wmma_reduce256(lds_pg, tid);
    const float tp  = wmma_reduce256(lds_p, tid);
    const float tg  = wmma_reduce256(lds_g, tid);
    const float td2 = wmma_reduce256(lds_d2, tid);
    if (tid == 0) {
      atomicAdd(acc_pg + slice, tpg);
      atomicAdd(acc_p + slice, tp);
      atomicAdd(acc_g + slice, tg);
      if (have_next) atomicAdd(acc_d2 + slice, td2);
    }
  }
}

// Single block: reduce 512 per-slice accumulators + mask logic -> scalar loss.
__global__ void __launch_bounds__(256) mdl_finalize_kernel(
    const int* __restrict__ mask, const float* __restrict__ acc_pg,
    const float* __restrict__ acc_p, const float* __restrict__ acc_g,
    const float* __restrict__ acc_d2, float* __restrict__ out) {
  __shared__ float s_inter, s_p, s_g, s_num;
  __shared__ int s_cnt;
  const int tid = threadIdx.x;
  if (tid == 0) {
    s_inter = 0.0f; s_p = 0.0f; s_g = 0.0f; s_num = 0.0f; s_cnt = 0;
  }
  __syncthreads();

  float inter = 0.0f, sp = 0.0f, sg = 0.0f, num = 0.0f;
  int cnt = 0;
  for (int i = tid; i < 512; i += 256) {
    const int ss = i & 63;
    const int mk = mask[i];
    const float mf = (mk == 1) ? 1.0f : 0.0f;
    inter += mf * acc_pg[i];
    sp    += mf * acc_p[i];
    sg    += mf * acc_g[i];
    if (ss < 63) {
      const bool a  = (mk == 1);
      const bool bb = (mask[i + 1] == 1);
      if (!(a && bb)) {  // contributes
        const float w = (a != bb) ? 1.0f : 0.0f;
        num += w * sqrtf(fmaxf(acc_d2[i], 0.0f));
        cnt += 1;
      }
    }
  }
  atomicAdd(&s_inter, inter);
  atomicAdd(&s_p, sp);
  atomicAdd(&s_g, sg);
  atomicAdd(&s_num, num);
  atomicAdd(&s_cnt, cnt);
  __syncthreads();

  if (tid == 0) {
    const float dice = (2.0f * s_inter + SMOOTH) / (s_p + s_g + SMOOTH);
    const float loss = 1.0f - dice;
    const float lc = (s_cnt > 0) ? (s_num / fmaxf((float)s_cnt, 1.0f)) : 0.0f;
    out[0] = loss + 0.001f * lc;
  }
}

extern "C" void kernel_launch(void* const* d_in, const int* in_sizes, int n_in,
                              void* d_out, int out_size, void* d_ws, size_t ws_size,
                              hipStream_t stream) {
  const float* pred = (const float*)d_in[0];  // [8,64,256,256] f32
  const float* gt   = (const float*)d_in[1];  // [8,64,256,256] f32
  const int*   mask = (const int*)d_in[2];    // [8,64] i32
  float* out = (float*)d_out;
  float* ws  = (float*)d_ws;

  float* acc_pg = ws;         // 512 floats
  float* acc_p  = ws + 512;   // 512 floats
  float* acc_g  = ws + 1024;  // 512 floats
  float* acc_d2 = ws + 1536;  // 512 floats (s=63 entries unused, stay zero)

  // Zero accumulators every call (harness does not re-poison between replays).
  mdl_init_kernel<<<8, 256, 0, stream>>>(ws);

  dim3 grid(32, 64, 8);  // chunks x slices x batches
  mdl_partial_kernel<<<grid, 256, 0, stream>>>(pred, gt, acc_pg, acc_p, acc_g,
                                               acc_d2);

  mdl_finalize_kernel<<<1, 256, 0, stream>>>(mask, acc_pg, acc_p, acc_g,
                                             acc_d2, out);
}